// CC3DTrack_61890478735941
// MI455X (gfx1250) — compile-verified
//
#include <hip/hip_runtime.h>
#include <hip/hip_bf16.h>

typedef float v2f __attribute__((ext_vector_type(2)));
typedef float v4f __attribute__((ext_vector_type(4)));
typedef float v8f __attribute__((ext_vector_type(8)));

static __device__ __forceinline__ int LT(int i, int j) { return i * (i + 1) / 2 + j; }

// ---------------------------------------------------------------------------
// Kalman update + box/velocity/acc bookkeeping. One lane per track.
// All small-matrix algebra fully unrolled so arrays live in VGPRs.
// ---------------------------------------------------------------------------
__global__ __launch_bounds__(128) void kf_update_kernel(
    const float* __restrict__ boxes,
    const float* __restrict__ obs,
    const float* __restrict__ prev_b7,
    const float* __restrict__ prev_means,
    const float* __restrict__ prev_covs,
    const float* __restrict__ prev_vel,
    const int* __restrict__ acc_frames,
    const int* __restrict__ frame_gap,
    const unsigned char* __restrict__ matched,
    const int* __restrict__ fps_p,
    float* __restrict__ o_boxes,
    float* __restrict__ o_means,
    float* __restrict__ o_covs,
    float* __restrict__ o_vel,
    float* __restrict__ o_acc,
    int N)
{
    int t = blockIdx.x * blockDim.x + threadIdx.x;
    if (t >= N) return;

    // ---- load P (10x10) with aligned float4 (stride 400B) ----
    float P[100];
    const v4f* pc = reinterpret_cast<const v4f*>(prev_covs + (size_t)t * 100);
    #pragma unroll
    for (int q = 0; q < 25; ++q) {
        v4f v = pc[q];
        P[4*q+0] = v.x; P[4*q+1] = v.y; P[4*q+2] = v.z; P[4*q+3] = v.w;
    }

    // ---- Cholesky of S = P[:7,:7] + I  (S = L L^T) ----
    float L[28];
    #pragma unroll
    for (int j = 0; j < 7; ++j) {
        float d = P[j*10+j] + 1.0f;
        #pragma unroll
        for (int k = 0; k < j; ++k) d -= L[LT(j,k)] * L[LT(j,k)];
        float dj = sqrtf(d);
        L[LT(j,j)] = dj;
        float inv = 1.0f / dj;
        #pragma unroll
        for (int i = j + 1; i < 7; ++i) {
            float s = P[i*10+j];
            #pragma unroll
            for (int k = 0; k < j; ++k) s -= L[LT(i,k)] * L[LT(j,k)];
            L[LT(i,j)] = s * inv;
        }
    }

    // ---- invert lower-triangular L -> Li ----
    float Li[28];
    #pragma unroll
    for (int j = 0; j < 7; ++j) {
        Li[LT(j,j)] = 1.0f / L[LT(j,j)];
        #pragma unroll
        for (int i = j + 1; i < 7; ++i) {
            float s = 0.0f;
            #pragma unroll
            for (int k = j; k < i; ++k) s += L[LT(i,k)] * Li[LT(k,j)];
            Li[LT(i,j)] = -s / L[LT(i,i)];
        }
    }

    // ---- G = PHt * Li^T (10x7); then G G^T = PHt S^-1 PHt^T = K S K^T ----
    float G[70];
    #pragma unroll
    for (int i = 0; i < 10; ++i) {
        #pragma unroll
        for (int j = 0; j < 7; ++j) {
            float s = 0.0f;
            #pragma unroll
            for (int k = 0; k <= j; ++k) s += P[i*10+k] * Li[LT(j,k)];
            G[i*7+j] = s;
        }
    }

    // ---- mean update: mu = m + G * (Li * innov) ----
    float m[10];
    const v2f* pm = reinterpret_cast<const v2f*>(prev_means + (size_t)t * 10);
    #pragma unroll
    for (int q = 0; q < 5; ++q) { v2f v = pm[q]; m[2*q] = v.x; m[2*q+1] = v.y; }

    float ob[7];
    #pragma unroll
    for (int k = 0; k < 7; ++k) ob[k] = obs[(size_t)t * 7 + k];

    float y[7];
    #pragma unroll
    for (int j = 0; j < 7; ++j) {
        float s = 0.0f;
        #pragma unroll
        for (int k = 0; k <= j; ++k) s += Li[LT(j,k)] * (ob[k] - m[k]);
        y[j] = s;
    }
    float mu[10];
    #pragma unroll
    for (int i = 0; i < 10; ++i) {
        float s = 0.0f;
        #pragma unroll
        for (int j = 0; j < 7; ++j) s += G[i*7+j] * y[j];
        mu[i] = m[i] + s;
    }

    const bool mt = matched[t] != 0;

    // ---- covariance output: matched ? P - G G^T : diag(10,...,10,1e4,1e4,1e4) ----
    v4f* oc = reinterpret_cast<v4f*>(o_covs + (size_t)t * 100);
    #pragma unroll
    for (int q = 0; q < 25; ++q) {
        float vv[4];
        #pragma unroll
        for (int e = 0; e < 4; ++e) {
            const int lin = 4*q + e;
            const int i = lin / 10, j = lin % 10;
            float cu = P[lin];
            #pragma unroll
            for (int k = 0; k < 7; ++k) cu -= G[i*7+k] * G[j*7+k];
            const float ci = (i == j) ? ((i < 7) ? 10.0f : 10000.0f) : 0.0f;
            vv[e] = mt ? cu : ci;
        }
        v4f o; o.x = vv[0]; o.y = vv[1]; o.z = vv[2]; o.w = vv[3];
        oc[q] = o;
    }

    // ---- means output ----
    v2f* om = reinterpret_cast<v2f*>(o_means + (size_t)t * 10);
    #pragma unroll
    for (int q = 0; q < 5; ++q) {
        const int i0 = 2*q, i1 = 2*q + 1;
        v2f o;
        o.x = mt ? mu[i0] : ((i0 < 7) ? ob[i0] : 0.0f);
        o.y = mt ? mu[i1] : ((i1 < 7) ? ob[i1] : 0.0f);
        om[q] = o;
    }

    // ---- boxes output ----
    float bx[12];
    const v4f* pb = reinterpret_cast<const v4f*>(boxes + (size_t)t * 12);
    #pragma unroll
    for (int q = 0; q < 3; ++q) {
        v4f v = pb[q];
        bx[4*q+0] = v.x; bx[4*q+1] = v.y; bx[4*q+2] = v.z; bx[4*q+3] = v.w;
    }
    const float fpsf = (float)fps_p[0];
    float nb[12];
    #pragma unroll
    for (int k = 0; k < 6; ++k) nb[k] = mt ? mu[k] : bx[k];
    nb[6] = bx[6];
    nb[7] = bx[7];
    nb[8] = mt ? mu[6] : bx[8];
    #pragma unroll
    for (int k = 0; k < 3; ++k) nb[9+k] = mt ? mu[7+k] * fpsf : bx[9+k];  // pred_loc - mu[:3] == mu[7:10]
    v4f* obx = reinterpret_cast<v4f*>(o_boxes + (size_t)t * 12);
    #pragma unroll
    for (int q = 0; q < 3; ++q) {
        v4f o; o.x = nb[4*q]; o.y = nb[4*q+1]; o.z = nb[4*q+2]; o.w = nb[4*q+3];
        obx[q] = o;
    }

    // ---- velocities & acc_frames ----
    const float accf = (float)acc_frames[t];
    const float gapf = (float)frame_gap[t];
    #pragma unroll
    for (int k = 0; k < 7; ++k) {
        float vo = (mu[k] - prev_b7[(size_t)t*7 + k]) / gapf;
        float vm = (prev_vel[(size_t)t*7 + k] * accf + vo) / (accf + 1.0f);
        o_vel[(size_t)t*7 + k] = mt ? vm : 0.0f;
    }
    o_acc[t] = mt ? accf + 1.0f : 0.0f;
}

// ---------------------------------------------------------------------------
// Embedding EMA via WMMA f32 16x16x4. One wave per 16 tracks.
// D = diag(wp)*Prev + diag(we)*Emb with wp = matched?0.2:0, we = matched?0.8:1
// (folds the matched-select into the weight diagonal; EXEC stays all-ones).
// ---------------------------------------------------------------------------
__global__ __launch_bounds__(32) void emb_ema_wmma_kernel(
    const float* __restrict__ prev_emb,
    const float* __restrict__ emb,
    const unsigned char* __restrict__ matched,
    float* __restrict__ o_emb)
{
    const int T    = blockIdx.x * 16;   // 16 tracks per wave
    const int lane = threadIdx.x;       // 0..31
    const int M    = lane & 15;         // A row (= D row base, = tile column n)
    const int kh   = lane >> 4;         // 0 -> K{0,1}, 1 -> K{2,3}
    const int kb   = kh * 2;

    const bool  mt = matched[T + M] != 0;
    const float wp = mt ? 0.2f : 0.0f;
    const float we = mt ? 0.8f : 1.0f;

    // A = K-slices of the 16x16 weight diagonal; invariant across column tiles.
    // A 16x4 f32 layout: lane M holds A[M][kb] in .x, A[M][kb+1] in .y.
    v2f ap[4], ae[4];
    #pragma unroll
    for (int j = 0; j < 4; ++j) {
        ap[j].x = (M == 4*j + kb    ) ? wp : 0.0f;
        ap[j].y = (M == 4*j + kb + 1) ? wp : 0.0f;
        ae[j].x = (M == 4*j + kb    ) ? we : 0.0f;
        ae[j].y = (M == 4*j + kb + 1) ? we : 0.0f;
    }

    #pragma unroll 4
    for (int c = 0; c < 256; c += 16) {
        v8f acc = {};
        #pragma unroll
        for (int j = 0; j < 4; ++j) {
            // B 4x16 f32 layout: lane n holds B[kb][n] in .x, B[kb+1][n] in .y.
            const int r = T + 4*j + kb;
            const size_t base = (size_t)r * 256 + c + M;
            v2f bp, be;
            bp.x = prev_emb[base];
            bp.y = prev_emb[base + 256];
            be.x = emb[base];
            be.y = emb[base + 256];
            acc = __builtin_amdgcn_wmma_f32_16x16x4_f32(
                false, ap[j], false, bp, (short)0, acc, false, false);
            acc = __builtin_amdgcn_wmma_f32_16x16x4_f32(
                false, ae[j], false, be, (short)0, acc, false, false);
        }
        // D 16x16 f32 layout: VGPR v = row v (lanes 0-15) / row v+8 (lanes 16-31).
        #pragma unroll
        for (int v = 0; v < 8; ++v) {
            const int row = T + v + kh * 8;
            o_emb[(size_t)row * 256 + c + M] = acc[v];
        }
    }
}

// ---------------------------------------------------------------------------
extern "C" void kernel_launch(void* const* d_in, const int* in_sizes, int n_in,
                              void* d_out, int out_size, void* d_ws, size_t ws_size,
                              hipStream_t stream)
{
    const int N = in_sizes[0] / 12;

    const float* boxes      = (const float*)d_in[0];
    const float* obs        = (const float*)d_in[1];
    const float* prev_b7    = (const float*)d_in[2];
    const float* prev_means = (const float*)d_in[3];
    const float* prev_covs  = (const float*)d_in[4];
    const float* prev_vel   = (const float*)d_in[5];
    const float* prev_emb   = (const float*)d_in[6];
    const float* emb        = (const float*)d_in[7];
    const int*   acc_frames = (const int*)d_in[8];
    const int*   frame_gap  = (const int*)d_in[9];
    const unsigned char* matched = (const unsigned char*)d_in[10];
    const int*   fps_p      = (const int*)d_in[11];

    float* out     = (float*)d_out;
    float* o_boxes = out;                       // N*12
    float* o_means = out + (size_t)N * 12;      // N*10
    float* o_covs  = out + (size_t)N * 22;      // N*100
    float* o_vel   = out + (size_t)N * 122;     // N*7
    float* o_emb   = out + (size_t)N * 129;     // N*256
    float* o_acc   = out + (size_t)N * 385;     // N*1

    kf_update_kernel<<<(N + 127) / 128, 128, 0, stream>>>(
        boxes, obs, prev_b7, prev_means, prev_covs, prev_vel,
        acc_frames, frame_gap, matched, fps_p,
        o_boxes, o_means, o_covs, o_vel, o_acc, N);

    emb_ema_wmma_kernel<<<N / 16, 32, 0, stream>>>(prev_emb, emb, matched, o_emb);
}